// DPPSearch_11012296147222
// MI455X (gfx1250) — compile-verified
//
#include <hip/hip_runtime.h>
#include <hip/hip_bf16.h>
#include <math.h>

// Problem constants (from reference)
#define NB    16
#define NL    64
#define NV    16384
#define DD    256
#define TOPK  16
#define NITER 8
#define EARLY 2
#define NROWS (NB * NL)           // 1024
#define K1    (2 * DD)            // 512  (GEMM1 reduction dim)

typedef __attribute__((ext_vector_type(2))) float v2f;
typedef __attribute__((ext_vector_type(8))) float v8f;

// -------- CDNA5 WMMA: D(16x16 f32) = A(16x4 f32) * B(4x16 f32) + C --------
__device__ __forceinline__ v8f wmma_f32_16x16x4(v2f a, v2f b, v8f c) {
    // 8-arg form: (neg_a, A, neg_b, B, c_mod, C, reuse_a, reuse_b)
    return __builtin_amdgcn_wmma_f32_16x16x4_f32(false, a, false, b, (short)0, c,
                                                 false, false);
}

// -------- CDNA5 async global->LDS copy (ASYNCcnt-tracked DMA path) --------
// LDS generic pointers carry the LDS byte address in addr[31:0] (ISA 10.2).
__device__ __forceinline__ unsigned lds_off(const void* p) {
    return (unsigned)(unsigned long long)(uintptr_t)p;
}
__device__ __forceinline__ void async_copy_b128(unsigned ldsOff, const void* gaddr) {
    asm volatile("global_load_async_to_lds_b128 %0, %1, off"
                 :: "v"(ldsOff), "v"((unsigned long long)(uintptr_t)gaddr)
                 : "memory");
}
__device__ __forceinline__ void wait_async_all() {
    asm volatile("s_wait_asynccnt 0x0" ::: "memory");
}

// ---------------- deterministic RNG (splitmix64 -> Gumbel) ----------------
__device__ __forceinline__ unsigned long long splitmix64(unsigned long long x) {
    x += 0x9E3779B97F4A7C15ull;
    x = (x ^ (x >> 30)) * 0xBF58476D1CE4E5B9ull;
    x = (x ^ (x >> 27)) * 0x94D049BB133111EBull;
    return x ^ (x >> 31);
}

// ---------------- per-thread sorted top-16 insertion ----------------------
__device__ __forceinline__ void tk_insert(float v, int id, float* lv, int* li) {
    if (v <= lv[TOPK - 1]) return;
    int j = TOPK - 1;
    while (j > 0 && lv[j - 1] < v) {
        lv[j] = lv[j - 1]; li[j] = li[j - 1]; --j;
    }
    lv[j] = v; li[j] = id;
}

// ===========================================================================
// Phase 1: per-row top-16 (values+indices), argmax (=top1 idx), row sum.
// One block (256 threads) per row of 16384 floats; float4 streaming loads.
// ===========================================================================
__global__ void topk_kernel(const float* __restrict__ probas,
                            float* __restrict__ tkv, int* __restrict__ tki,
                            int* __restrict__ amax, float* __restrict__ rowsum) {
    const int row = blockIdx.x;
    const float4* p4 = (const float4*)(probas + (size_t)row * NV);

    float lv[TOPK]; int li[TOPK];
    #pragma unroll
    for (int i = 0; i < TOPK; ++i) { lv[i] = -3.0e38f; li[i] = 0; }

    float s = 0.f;
    for (int c = threadIdx.x; c < NV / 4; c += 256) {
        __builtin_prefetch(&p4[c + 256], 0, 0);    // global_prefetch_b8
        float4 v4 = p4[c];
        const int v = c << 2;
        s += v4.x + v4.y + v4.z + v4.w;
        tk_insert(v4.x, v + 0, lv, li);
        tk_insert(v4.y, v + 1, lv, li);
        tk_insert(v4.z, v + 2, lv, li);
        tk_insert(v4.w, v + 3, lv, li);
    }

    __shared__ float sv[256 * TOPK];
    __shared__ int   si[256 * TOPK];
    __shared__ float ssum[256];
    #pragma unroll
    for (int i = 0; i < TOPK; ++i) {
        sv[threadIdx.x * TOPK + i] = lv[i];
        si[threadIdx.x * TOPK + i] = li[i];
    }
    ssum[threadIdx.x] = s;
    __syncthreads();

    // stage 1: 16 mergers, each folds 16 threads' lists (256 candidates)
    if (threadIdx.x < 16) {
        float mv[TOPK]; int mi[TOPK];
        #pragma unroll
        for (int i = 0; i < TOPK; ++i) { mv[i] = -3.0e38f; mi[i] = 0; }
        const int base = threadIdx.x * 16;
        for (int t = 0; t < 16; ++t)
            for (int i = 0; i < TOPK; ++i)
                tk_insert(sv[(base + t) * TOPK + i], si[(base + t) * TOPK + i], mv, mi);
        #pragma unroll
        for (int i = 0; i < TOPK; ++i) {
            sv[threadIdx.x * TOPK + i] = mv[i];
            si[threadIdx.x * TOPK + i] = mi[i];
        }
    }
    __syncthreads();

    // stage 2: final fold of 256 candidates + sum reduction
    if (threadIdx.x == 0) {
        float mv[TOPK]; int mi[TOPK];
        #pragma unroll
        for (int i = 0; i < TOPK; ++i) { mv[i] = -3.0e38f; mi[i] = 0; }
        for (int t = 0; t < 16; ++t)
            for (int i = 0; i < TOPK; ++i)
                tk_insert(sv[t * TOPK + i], si[t * TOPK + i], mv, mi);
        #pragma unroll
        for (int i = 0; i < TOPK; ++i) {
            tkv[row * TOPK + i] = mv[i];
            tki[row * TOPK + i] = mi[i];
        }
        amax[row] = mi[0];
        float tot = 0.f;
        for (int t = 0; t < 256; ++t) tot += ssum[t];
        rowsum[row] = tot;
    }
}

// ===========================================================================
// Phase 2a: Gumbel-argmax categorical sample over 16 top-k logits per (b,l)
// ===========================================================================
__global__ void sample_kernel(const float* __restrict__ tkv,
                              const int* __restrict__ tki,
                              const int* __restrict__ amax,
                              const int* __restrict__ mask,
                              int* __restrict__ samples, int t) {
    const int row = blockIdx.x * blockDim.x + threadIdx.x;
    if (row >= NROWS) return;
    const int b = row >> 6, l = row & 63;
    const int m = mask[row];

    int slen = 0;
    for (int j = 0; j < NL; ++j) slen += (mask[b * NL + j] != 0);

    float bestv = -3.0e38f; int bestk = 0;
    for (int k = 0; k < TOPK; ++k) {
        float logit = m ? logf(tkv[row * TOPK + k]) : 0.0f;
        unsigned long long h =
            splitmix64(((unsigned long long)(t * NROWS + row) << 8) |
                       (unsigned long long)k | 0xC0FFEE0000000000ull);
        float u = ((float)(h >> 40) + 0.5f) * (1.0f / 16777216.0f);
        float g = -logf(-logf(u));
        float sc = logit + g;
        if (sc > bestv) { bestv = sc; bestk = k; }
    }
    int samp = tki[row * TOPK + bestk];
    if (l == slen - 1) samp = amax[row];     // one_hot_by_mask position -> MAP
    samples[t * NROWS + row] = samp;
}

// ===========================================================================
// Phase 2b: X[row] = [ emb_table[batch_vocab[sample]] , h_d[row] ] * maskf
// ===========================================================================
__global__ void build_x_kernel(const int* __restrict__ samples,
                               const int* __restrict__ batch_vocab,
                               const float* __restrict__ emb_table,
                               const float* __restrict__ h_d,
                               const int* __restrict__ mask,
                               float* __restrict__ X, int t) {
    const int row = blockIdx.x;
    const int samp = samples[t * NROWS + row];
    const float mf = mask[row] ? 1.0f : 0.0f;
    const int tok = batch_vocab[samp];
    const float* er = emb_table + (size_t)tok * DD;
    const float* hr = h_d + (size_t)row * DD;
    float* xr = X + (size_t)row * K1;
    for (int c = threadIdx.x; c < DD; c += blockDim.x) {
        xr[c]      = er[c] * mf;
        xr[DD + c] = hr[c] * mf;
    }
}

// ===========================================================================
// Phase 2c: E = ReLU(X @ W1 + b1)   [1024,512]x[512,256]
// One block per 16-row M-slab: async-stage the 16x512 X slab (32KB) into LDS,
// then 8 waves x 2 N-tiles of V_WMMA_F32_16X16X4_F32 with A-frags from LDS.
// W1 stays in L2 (coalesced along N).
// ===========================================================================
__global__ void gemm1_kernel(const float* __restrict__ X,
                             const float* __restrict__ W1,
                             const float* __restrict__ b1,
                             float* __restrict__ E) {
    __shared__ float sX[16 * K1];                    // 32 KB
    const int tileM = blockIdx.x * 16;

    // async global->LDS stage: 2048 x 16B chunks, 256 threads x 8 issues
    {
        const char* src = (const char*)(X + (size_t)tileM * K1);
        char* dst = (char*)sX;
        #pragma unroll
        for (int i = 0; i < 8; ++i) {
            const int chunk = i * 256 + threadIdx.x;
            async_copy_b128(lds_off(dst + chunk * 16), src + chunk * 16);
        }
        wait_async_all();
        __syncthreads();
    }

    const int wave = threadIdx.x >> 5;               // 0..7
    const int lane = threadIdx.x & 31;
    const int half = lane >> 4;                      // 0: K+{0,1}  1: K+{2,3}
    const int l16  = lane & 15;

    #pragma unroll
    for (int j = 0; j < 2; ++j) {
        const int tileN = (wave * 2 + j) * 16;
        const int n = tileN + l16;
        v8f c = {};
        for (int k = 0; k < K1; k += 4) {
            const int ka = k + half * 2;
            v2f a;  a.x = sX[l16 * K1 + ka];         // ds_load_b64
                    a.y = sX[l16 * K1 + ka + 1];
            v2f bf; bf.x = W1[(size_t)ka * DD + n];
                    bf.y = W1[(size_t)(ka + 1) * DD + n];
            c = wmma_f32_16x16x4(a, bf, c);
        }
        const float bias = b1[n];
        #pragma unroll
        for (int r = 0; r < 8; ++r) {
            const int mr = tileM + r + half * 8;     // C/D layout: half selects M+8
            const float v = c[r] + bias;
            E[(size_t)mr * DD + tileN + l16] = v > 0.f ? v : 0.f;
        }
    }
}

// ===========================================================================
// Phase 2d: per-batch Gram  K_b = E_b @ E_b^T   [64,256]x[256,64]
// One block per batch: async-stage E_b (64KB) into LDS; 8 waves x 2 tiles,
// both A and B(=E^T) fragments read from LDS.
// ===========================================================================
__global__ void gram_kernel(const float* __restrict__ E, float* __restrict__ Km) {
    __shared__ float sE[NL * DD];                    // 64 KB
    const int bat = blockIdx.x;

    {
        const char* src = (const char*)(E + (size_t)bat * NL * DD);
        char* dst = (char*)sE;
        #pragma unroll
        for (int i = 0; i < 16; ++i) {               // 4096 x 16B chunks
            const int chunk = i * 256 + threadIdx.x;
            async_copy_b128(lds_off(dst + chunk * 16), src + chunk * 16);
        }
        wait_async_all();
        __syncthreads();
    }

    const int wave = threadIdx.x >> 5;
    const int lane = threadIdx.x & 31;
    const int half = lane >> 4;
    const int l16  = lane & 15;
    float* Kb = Km + (size_t)bat * NL * NL;

    #pragma unroll
    for (int j = 0; j < 2; ++j) {
        const int tile = wave * 2 + j;               // 0..15 -> 4x4 tile grid
        const int tileM = (tile >> 2) << 4;
        const int tileN = (tile & 3) << 4;
        v8f c = {};
        for (int k = 0; k < DD; k += 4) {
            const int ka = k + half * 2;
            v2f a;  a.x = sE[(tileM + l16) * DD + ka];
                    a.y = sE[(tileM + l16) * DD + ka + 1];
            v2f bf; bf.x = sE[(tileN + l16) * DD + ka];   // B = E^T
                    bf.y = sE[(tileN + l16) * DD + ka + 1];
            c = wmma_f32_16x16x4(a, bf, c);
        }
        #pragma unroll
        for (int r = 0; r < 8; ++r)
            Kb[(size_t)(tileM + r + half * 8) * NL + tileN + l16] = c[r];
    }
}

// ===========================================================================
// Phase 2e: score[b] = det( where(m2d, K_b, I) )  via LDS Gaussian elimination
// (Gram matrices are PSD -> no pivoting). One block of 64 threads per batch.
// ===========================================================================
__global__ void det_kernel(const float* __restrict__ Km,
                           const int* __restrict__ mask,
                           float* __restrict__ scores, int t) {
    __shared__ float A[NL][NL + 1];
    const int b = blockIdx.x;
    const int r = threadIdx.x;                  // 64 threads, one row each
    const int mr = mask[b * NL + r];
    for (int cI = 0; cI < NL; ++cI) {
        const int mc = mask[b * NL + cI];
        const float kv = Km[(size_t)b * NL * NL + r * NL + cI];
        const float ev = (r == cI) ? 1.f : 0.f;
        A[r][cI] = (mr && mc) ? kv : ev;
    }
    __syncthreads();
    for (int p = 0; p < NL; ++p) {
        const float piv = A[p][p];
        if (r > p) {
            const float f = A[r][p] / piv;
            for (int cI = p; cI < NL; ++cI) A[r][cI] -= f * A[p][cI];
        }
        __syncthreads();
    }
    if (r == 0) {
        float det = 1.f;
        for (int i = 0; i < NL; ++i) det *= A[i][i];
        scores[t * NB + b] = det;
    }
}

// ===========================================================================
// Phase 2f: replay the scan carry (improved/count/stopped) over the 8 stored
// score vectors; emit best[row] and max_score (appended after the big output).
// ===========================================================================
__global__ void select_kernel(const float* __restrict__ scores,
                              const int* __restrict__ samples,
                              const int* __restrict__ amax,
                              int* __restrict__ best,
                              float* __restrict__ out_maxscore) {
    __shared__ int   best_iter[NB];
    __shared__ float ms[NB];
    if (threadIdx.x == 0) {
        float mxs[NB]; int bi[NB];
        for (int b = 0; b < NB; ++b) { mxs[b] = -3.0e38f; bi[b] = -1; }
        int count = 0; bool stopped = false;
        for (int t = 0; t < NITER; ++t) {
            bool any = false; bool imp[NB];
            for (int b = 0; b < NB; ++b) {
                imp[b] = scores[t * NB + b] > mxs[b];
                any = any || imp[b];
            }
            count = any ? 0 : (count + 1);
            for (int b = 0; b < NB; ++b)
                if (imp[b] && !stopped) { mxs[b] = scores[t * NB + b]; bi[b] = t; }
            stopped = stopped || ((!any) && (count >= EARLY));
        }
        for (int b = 0; b < NB; ++b) { best_iter[b] = bi[b]; ms[b] = mxs[b]; }
    }
    __syncthreads();
    for (int row = threadIdx.x; row < NROWS; row += blockDim.x) {
        const int bi = best_iter[row >> 6];
        best[row] = (bi >= 0) ? samples[bi * NROWS + row] : amax[row];
    }
    if (threadIdx.x < NB) out_maxscore[threadIdx.x] = ms[threadIdx.x];
}

// ===========================================================================
// Phase 3: analytic renormalizer  nm = 0.2*rowsum + 0.6*p_best  (or 1e-10)
// then out = probas * (v==best ? 0.8 : 0.2) * (1/nm) -- single float4 pass.
// ===========================================================================
__global__ void nm_kernel(const float* __restrict__ probas,
                          const float* __restrict__ rowsum,
                          const int* __restrict__ mask,
                          const int* __restrict__ best,
                          float* __restrict__ invnm) {
    const int row = blockIdx.x * blockDim.x + threadIdx.x;
    if (row >= NROWS) return;
    const float pb = probas[(size_t)row * NV + best[row]];
    const float nm = mask[row] ? (0.2f * rowsum[row] + 0.6f * pb) : 1e-10f;
    invnm[row] = 1.0f / nm;
}

__global__ void finalize_kernel(const float4* __restrict__ probas4,
                                const int* __restrict__ best,
                                const float* __restrict__ invnm,
                                float4* __restrict__ out4) {
    const size_t idx = (size_t)blockIdx.x * blockDim.x + threadIdx.x;  // NV/4=4096/row
    const int row = (int)(idx >> 12);
    const int v0 = (int)((idx & 4095) << 2);
    const int bv = best[row];
    const float inm = invnm[row];
    float4 p = probas4[idx];
    p.x *= ((v0 + 0) == bv ? 0.8f : 0.2f) * inm;
    p.y *= ((v0 + 1) == bv ? 0.8f : 0.2f) * inm;
    p.z *= ((v0 + 2) == bv ? 0.8f : 0.2f) * inm;
    p.w *= ((v0 + 3) == bv ? 0.8f : 0.2f) * inm;
    out4[idx] = p;
}

// ===========================================================================
extern "C" void kernel_launch(void* const* d_in, const int* in_sizes, int n_in,
                              void* d_out, int out_size, void* d_ws, size_t ws_size,
                              hipStream_t stream) {
    const float* probas      = (const float*)d_in[0];
    const float* h_d         = (const float*)d_in[1];
    const int*   mask        = (const int*)d_in[2];
    const int*   batch_vocab = (const int*)d_in[3];
    const float* emb_table   = (const float*)d_in[4];
    const float* W1          = (const float*)d_in[5];
    const float* b1          = (const float*)d_in[6];
    float* out = (float*)d_out;

    // workspace carve-up (all 256B-aligned chunks)
    char* ws = (char*)d_ws;
    size_t off = 0;
    auto carve = [&](size_t bytes) {
        void* p = ws + off;
        off += (bytes + 255) & ~(size_t)255;
        return p;
    };
    float* tkv     = (float*)carve(NROWS * TOPK * sizeof(float));
    int*   tki     = (int*)  carve(NROWS * TOPK * sizeof(int));
    int*   amax    = (int*)  carve(NROWS * sizeof(int));
    float* rowsum  = (float*)carve(NROWS * sizeof(float));
    int*   samples = (int*)  carve(NITER * NROWS * sizeof(int));
    float* scores  = (float*)carve(NITER * NB * sizeof(float));
    float* X       = (float*)carve((size_t)NROWS * K1 * sizeof(float));
    float* E       = (float*)carve((size_t)NROWS * DD * sizeof(float));
    float* Km      = (float*)carve((size_t)NB * NL * NL * sizeof(float));
    int*   best    = (int*)  carve(NROWS * sizeof(int));
    float* invnm   = (float*)carve(NROWS * sizeof(float));
    (void)ws_size; (void)in_sizes; (void)n_in; (void)out_size;

    // Phase 1: top-k / argmax / rowsum (one float4 pass over probas)
    topk_kernel<<<NROWS, 256, 0, stream>>>(probas, tkv, tki, amax, rowsum);

    // Phase 2: 8 search iterations (fixed kernel sequence; carry replayed later)
    for (int t = 0; t < NITER; ++t) {
        sample_kernel<<<(NROWS + 255) / 256, 256, 0, stream>>>(tkv, tki, amax, mask,
                                                               samples, t);
        build_x_kernel<<<NROWS, 256, 0, stream>>>(samples, batch_vocab, emb_table,
                                                  h_d, mask, X, t);
        gemm1_kernel<<<NROWS / 16, 256, 0, stream>>>(X, W1, b1, E);  // 64 blocks
        gram_kernel<<<NB, 256, 0, stream>>>(E, Km);                  // 16 blocks
        det_kernel<<<NB, NL, 0, stream>>>(Km, mask, scores, t);
    }

    // Phase 2f: scan replay -> best tokens + max_score tail of d_out
    select_kernel<<<1, 256, 0, stream>>>(scores, samples, amax, best,
                                         out + (size_t)NB * NL * NV);

    // Phase 3: analytic renorm + single float4 pass over probas
    nm_kernel<<<(NROWS + 255) / 256, 256, 0, stream>>>(probas, rowsum, mask, best,
                                                       invnm);
    finalize_kernel<<<(size_t)NB * NL * NV / 4 / 256, 256, 0, stream>>>(
        (const float4*)probas, best, invnm, (float4*)out);
}